// DynamicCache_61151744361024
// MI455X (gfx1250) — compile-verified
//
#include <hip/hip_runtime.h>
#include <stddef.h>

// Problem shape (fixed by the reference)
constexpr int B  = 4;
constexpr int H  = 32;
constexpr int SP = 4096;   // past sequence length
constexpr int SN = 16;     // new tokens
constexpr int ST = SP + SN;  // 4112
constexpr int D  = 128;
constexpr int BH = B * H;    // 128

constexpr size_t KV_ELEMS = (size_t)BH * ST * D;   // per-tensor output elements (K or V)

typedef float v4f __attribute__((ext_vector_type(4)));

constexpr int TPB   = 256;  // 8 wave32 per block
constexpr int TILES = 8;    // async tiles per thread -> 4KB in flight per wave per phase

// Flat (generic) address of a __shared__ object: bits [31:0] are the
// wave-relative LDS byte offset (ISA 10.2 aperture mapping).
__device__ __forceinline__ unsigned lds_byte_off(const void* p) {
  return (unsigned)(unsigned long long)p;
}

// Bulk past-K/V copy via the CDNA5 async global<->LDS datapath.
// grid = (64, BH, 2): x covers 2MB/bh in 32KB chunks, y = bh, z = K/V.
__global__ void __launch_bounds__(TPB)
past_concat_async(const float* __restrict__ past_k,
                  const float* __restrict__ past_v,
                  float* __restrict__ out) {
  __shared__ v4f stage[TPB * TILES];   // 32 KB staging buffer

  const float* src  = (blockIdx.z == 0) ? past_k : past_v;
  float*       dstT = out + (size_t)blockIdx.z * KV_ELEMS;

  const size_t bh = blockIdx.y;
  const v4f* __restrict__ gsrc = (const v4f*)(src  + bh * (size_t)SP * D);
  v4f* __restrict__       gdst = (v4f*)(dstT + bh * (size_t)ST * D);

  const size_t   base = (size_t)blockIdx.x * (TPB * TILES);
  const unsigned tid  = threadIdx.x;

  // Phase 1: issue TILES async HBM->LDS loads per lane (ASYNCcnt batching).
#pragma unroll
  for (int i = 0; i < TILES; ++i) {
    const unsigned slot = (unsigned)i * TPB + tid;
    const unsigned l    = lds_byte_off(&stage[slot]);
    const v4f*     g    = gsrc + base + slot;
    asm volatile("global_load_async_to_lds_b128 %0, %1, off th:TH_LOAD_NT"
                 :: "v"(l), "v"(g) : "memory");
  }
  asm volatile("s_wait_asynccnt 0x0" ::: "memory");

  // Phase 2: issue TILES async LDS->HBM stores per lane.
#pragma unroll
  for (int i = 0; i < TILES; ++i) {
    const unsigned slot = (unsigned)i * TPB + tid;
    const unsigned l    = lds_byte_off(&stage[slot]);
    v4f*           g    = gdst + base + slot;
    asm volatile("global_store_async_from_lds_b128 %0, %1, off th:TH_STORE_NT"
                 :: "v"(g), "v"(l) : "memory");
  }
  asm volatile("s_wait_asynccnt 0x0" ::: "memory");
}

// Append the 16 new tokens per (b,h): tiny (2 MB total), direct B128 NT copy.
// grid = (2, BH, 2): x*256 threads cover SN*D/4 = 512 float4 per bh.
__global__ void __launch_bounds__(TPB)
new_token_append(const float* __restrict__ key_states,
                 const float* __restrict__ value_states,
                 float* __restrict__ out) {
  const float* src  = (blockIdx.z == 0) ? key_states : value_states;
  float*       dstT = out + (size_t)blockIdx.z * KV_ELEMS;

  const size_t   bh  = blockIdx.y;
  const unsigned idx = blockIdx.x * TPB + threadIdx.x;   // 0..511

  const v4f* s4 = (const v4f*)(src + bh * (size_t)SN * D);
  v4f*       d4 = (v4f*)(dstT + bh * (size_t)ST * D + (size_t)SP * D);

  v4f v = __builtin_nontemporal_load(s4 + idx);
  __builtin_nontemporal_store(v, d4 + idx);
}

extern "C" void kernel_launch(void* const* d_in, const int* in_sizes, int n_in,
                              void* d_out, int out_size, void* d_ws, size_t ws_size,
                              hipStream_t stream) {
  (void)in_sizes; (void)n_in; (void)out_size; (void)d_ws; (void)ws_size;

  const float* past_k = (const float*)d_in[0];
  const float* past_v = (const float*)d_in[1];
  const float* key_s  = (const float*)d_in[2];
  const float* val_s  = (const float*)d_in[3];
  float* out = (float*)d_out;

  // Past copy: per (bh): SP*D floats = 2MB = 64 blocks * (256 thr * 8 tiles * 16B)
  dim3 gPast(64, BH, 2);
  past_concat_async<<<gPast, TPB, 0, stream>>>(past_k, past_v, out);

  // New-token append: per (bh): SN*D/4 = 512 float4 = 2 blocks * 256 threads
  dim3 gNew(2, BH, 2);
  new_token_append<<<gNew, TPB, 0, stream>>>(key_s, val_s, out);
}